// ConvOut_35029753266847
// MI455X (gfx1250) — compile-verified
//
#include <hip/hip_runtime.h>

typedef __attribute__((ext_vector_type(2))) float v2f;
typedef __attribute__((ext_vector_type(8))) float v8f;

#define ETA 0.005f

// Async global->LDS copy of one 32-bit element per lane (gfx1250 ASYNCcnt path).
__device__ __forceinline__ void async_g2l_b32(unsigned lds_addr, unsigned voff_bytes,
                                              const void* sbase) {
  asm volatile("global_load_async_to_lds_b32 %0, %1, %2 offset:0"
               :: "v"(lds_addr), "v"(voff_bytes), "s"(sbase) : "memory");
}
__device__ __forceinline__ void wait_async0() {
  asm volatile("s_wait_asynccnt 0" ::: "memory");
}

// ---------------------------------------------------------------------------
// Kernel 1: y[b] = W @ x[b] @ W^T, via two f32 WMMA GEMMs per batch element.
// One block (256 thr = 8 wave32) per batch element; operands staged in LDS via
// async global->LDS copies (bypass VGPRs), compute on v_wmma_f32_16x16x4_f32.
// ---------------------------------------------------------------------------
__global__ __launch_bounds__(256) void bimap_kernel(const float* __restrict__ x,
                                                    const float* __restrict__ W,
                                                    float* __restrict__ y) {
  __shared__ float xs[64][65];  // x[b]
  __shared__ float wm[64][65];  // W
  __shared__ float ts[64][65];  // t = x @ W^T
  const int t = threadIdx.x;
  const int b = blockIdx.x;
  const float* xb = x + (size_t)b * 4096;

  // Async-stage x[b] and W into LDS (stride-65 padded to dodge bank conflicts).
  {
    unsigned lds_xs = (unsigned)(uintptr_t)&xs[0][0];
    unsigned lds_wm = (unsigned)(uintptr_t)&wm[0][0];
    for (int i = 0; i < 16; ++i) {
      int idx = t + 256 * i;
      int row = idx >> 6, col = idx & 63;
      unsigned d = (unsigned)(row * 65 + col) * 4u;
      unsigned voff = (unsigned)idx * 4u;
      async_g2l_b32(lds_xs + d, voff, xb);
      async_g2l_b32(lds_wm + d, voff, W);
    }
  }
  wait_async0();
  __syncthreads();

  const int wave = t >> 5;
  const int lane = t & 31;
  const int lo = lane & 15;
  const int hi = lane >> 4;

  // GEMM1: t = x @ W^T   (B[l][j] = W[j][l])
  for (int tt = wave; tt < 16; tt += 8) {
    int ti = tt >> 2, tj = tt & 3;
    int m = ti * 16 + lo;
    int n = tj * 16 + lo;
    v8f c = {};
    for (int k = 0; k < 64; k += 4) {
      int kb = k + (hi ? 2 : 0);
      v2f a;  a.x = xs[m][kb];  a.y = xs[m][kb + 1];
      v2f bb; bb.x = wm[n][kb]; bb.y = wm[n][kb + 1];
      c = __builtin_amdgcn_wmma_f32_16x16x4_f32(false, a, false, bb,
                                                (short)0, c, false, false);
    }
    for (int r = 0; r < 8; ++r)
      ts[ti * 16 + r + (hi ? 8 : 0)][tj * 16 + lo] = c[r];
  }
  __syncthreads();

  // GEMM2: y = W @ t
  float* yb = y + (size_t)b * 4096;
  for (int tt = wave; tt < 16; tt += 8) {
    int ti = tt >> 2, tj = tt & 3;
    int m = ti * 16 + lo;
    int n = tj * 16 + lo;
    v8f c = {};
    for (int k = 0; k < 64; k += 4) {
      int kb = k + (hi ? 2 : 0);
      v2f a;  a.x = wm[m][kb];     a.y = wm[m][kb + 1];
      v2f bb; bb.x = ts[kb][n];    bb.y = ts[kb + 1][n];
      c = __builtin_amdgcn_wmma_f32_16x16x4_f32(false, a, false, bb,
                                                (short)0, c, false, false);
    }
    for (int r = 0; r < 8; ++r)
      yb[(ti * 16 + r + (hi ? 8 : 0)) * 64 + tj * 16 + lo] = c[r];
  }
}

// ---------------------------------------------------------------------------
// Kernel 2: per-(i,j) batch mean and unbiased (ddof=1) inverse std.
// 16 blocks x 256 threads: block covers 256 contiguous positions, streams
// the batch axis coalesced, with global prefetch hints.
// ---------------------------------------------------------------------------
__global__ __launch_bounds__(256) void stats_kernel(const float* __restrict__ y,
                                                    float* __restrict__ mean,
                                                    float* __restrict__ rstd) {
  const int pos = blockIdx.x * 256 + threadIdx.x;  // 0..4095
  const float* p = y + pos;
  float s = 0.f, sq = 0.f;
  for (int b = 0; b < 8192; ++b) {
    if (b + 16 < 8192) __builtin_prefetch(p + (size_t)(b + 16) * 4096, 0, 0);
    float v = p[(size_t)b * 4096];
    s += v;
    sq += v * v;
  }
  float m = s * (1.0f / 8192.0f);
  float var = (sq - 8192.0f * m * m) * (1.0f / 8191.0f);
  var = fmaxf(var, 1e-20f);
  mean[pos] = m;
  rstd[pos] = rsqrtf(var);
}

// ---------------------------------------------------------------------------
// Kernel 3: normalize + symmetrize (lower triangle), batched two-sided cyclic
// Jacobi eigensolver (9 sweeps, 63 chess-tournament rounds of 32 disjoint
// rotations), eigenvalue threshold at ETA, reconstruct V*diag(lam)*V^T via
// f32 WMMA. Operates in place on d_out: async-stage y[b] into LDS (reusing
// V's storage as scratch) before any write-back.
// ---------------------------------------------------------------------------
__global__ __launch_bounds__(256) void eig_kernel(const float* __restrict__ mean,
                                                  const float* __restrict__ rstd,
                                                  float* __restrict__ out) {
  __shared__ float A[64][65];
  __shared__ float V[64][65];
  __shared__ float rc[32], rs[32];
  __shared__ int pp[32], qq[32];
  __shared__ float lam[64];
  const int t = threadIdx.x;
  const int b = blockIdx.x;
  float* yb = out + (size_t)b * 4096;

  // Async-stage raw y[b] into V[][] (scratch) -- single coalesced pass.
  {
    unsigned lds_v = (unsigned)(uintptr_t)&V[0][0];
    for (int i = 0; i < 16; ++i) {
      int idx = t + 256 * i;
      int row = idx >> 6, col = idx & 63;
      unsigned d = (unsigned)(row * 65 + col) * 4u;
      async_g2l_b32(lds_v + d, (unsigned)idx * 4u, yb);
    }
  }
  wait_async0();
  __syncthreads();

  // Build normalized lower-triangle symmetric A from staged y.
  for (int i = 0; i < 16; ++i) {
    int idx = t + 256 * i;
    int row = idx >> 6, col = idx & 63;
    int sr = (row >= col) ? row : col;
    int sc = (row >= col) ? col : row;
    int sp = sr * 64 + sc;
    A[row][col] = (V[sr][sc] - mean[sp]) * rstd[sp];
  }
  __syncthreads();
  // Now initialize V = I.
  for (int i = 0; i < 16; ++i) {
    int idx = t + 256 * i;
    int row = idx >> 6, col = idx & 63;
    V[row][col] = (row == col) ? 1.0f : 0.0f;
  }
  __syncthreads();

  for (int sweep = 0; sweep < 9; ++sweep) {
    for (int r = 0; r < 63; ++r) {
      // 32 disjoint Jacobi rotations (round-robin tournament pairing).
      if (t < 32) {
        int p, q;
        if (t == 0) { p = 63; q = r % 63; }
        else {
          p = (r + t) % 63;
          q = (r - t + 63) % 63;
        }
        if (p > q) { int tmp = p; p = q; q = tmp; }
        pp[t] = p; qq[t] = q;
        float app = A[p][p], aqq = A[q][q], apq = A[p][q];
        float c = 1.0f, s = 0.0f;
        if (fabsf(apq) > 1e-12f) {
          float tau = (aqq - app) / (2.0f * apq);
          float th = (tau >= 0.f ? 1.0f : -1.0f) /
                     (fabsf(tau) + sqrtf(1.0f + tau * tau));
          c = rsqrtf(1.0f + th * th);
          s = th * c;
        }
        rc[t] = c; rs[t] = s;
      }
      __syncthreads();
      // Column rotations: A <- A*J and V <- V*J (32 pairs x 64 rows).
      for (int i = 0; i < 8; ++i) {
        int idx = t + 256 * i;
        int k = idx >> 6, rr = idx & 63;
        int p = pp[k], q = qq[k];
        float c = rc[k], s = rs[k];
        float ap = A[rr][p], aq = A[rr][q];
        A[rr][p] = c * ap - s * aq;
        A[rr][q] = s * ap + c * aq;
        float vp = V[rr][p], vq = V[rr][q];
        V[rr][p] = c * vp - s * vq;
        V[rr][q] = s * vp + c * vq;
      }
      __syncthreads();
      // Row rotations: A <- J^T * A (32 pairs x 64 cols).
      for (int i = 0; i < 8; ++i) {
        int idx = t + 256 * i;
        int k = idx >> 6, cc = idx & 63;
        int p = pp[k], q = qq[k];
        float c = rc[k], s = rs[k];
        float ap = A[p][cc], aq = A[q][cc];
        A[p][cc] = c * ap - s * aq;
        A[q][cc] = s * ap + c * aq;
      }
      __syncthreads();
    }
  }

  if (t < 64) lam[t] = fmaxf(A[t][t], ETA);
  __syncthreads();

  // out[b] = (V * diag(lam)) @ V^T via f32 WMMA.
  const int wave = t >> 5;
  const int lane = t & 31;
  const int lo = lane & 15;
  const int hi = lane >> 4;
  for (int tt = wave; tt < 16; tt += 8) {
    int ti = tt >> 2, tj = tt & 3;
    int m = ti * 16 + lo;
    int n = tj * 16 + lo;
    v8f c = {};
    for (int k = 0; k < 64; k += 4) {
      int kb = k + (hi ? 2 : 0);
      v2f a;  a.x = V[m][kb] * lam[kb];  a.y = V[m][kb + 1] * lam[kb + 1];
      v2f bb; bb.x = V[n][kb];           bb.y = V[n][kb + 1];
      c = __builtin_amdgcn_wmma_f32_16x16x4_f32(false, a, false, bb,
                                                (short)0, c, false, false);
    }
    for (int r2 = 0; r2 < 8; ++r2)
      yb[(ti * 16 + r2 + (hi ? 8 : 0)) * 64 + tj * 16 + lo] = c[r2];
  }
}

// ---------------------------------------------------------------------------
extern "C" void kernel_launch(void* const* d_in, const int* in_sizes, int n_in,
                              void* d_out, int out_size, void* d_ws, size_t ws_size,
                              hipStream_t stream) {
  const float* x = (const float*)d_in[0];  // [8192,64,64]
  const float* W = (const float*)d_in[1];  // [64,64]
  float* out = (float*)d_out;              // [8192,64,64]
  float* mean = (float*)d_ws;              // 4096 floats
  float* rstd = mean + 4096;               // 4096 floats

  bimap_kernel<<<8192, 256, 0, stream>>>(x, W, out);      // y -> d_out
  stats_kernel<<<16, 256, 0, stream>>>(out, mean, rstd);  // batch stats
  eig_kernel<<<8192, 256, 0, stream>>>(mean, rstd, out);  // in-place eig+recon
}